// LUTCompatibility_43327630082122
// MI455X (gfx1250) — compile-verified
//
#include <hip/hip_runtime.h>
#include <math.h>

#define NBX 168
#define NBY 480
#define NBL 6
#define EXT 2
#define W   5
#define INV_SQRT2 0.70710678118654752440f
#define CELLS (NBX * NBY)        // 80640
#define DEM_N (CELLS * NBL)      // 483840
#define DEM_PAD 8                // padding so A-matrix k=6,7 loads stay in-bounds

typedef __attribute__((ext_vector_type(2))) float v2f;
typedef __attribute__((ext_vector_type(8))) float v8f;

// ---------------------------------------------------------------------------
// Shared per-axis truncated-Gaussian demand: 6 erf evals, 5 bin weights.
// ---------------------------------------------------------------------------
__device__ __forceinline__ void axis_demand(float c, int nb, float g[W], int bc[W]) {
    int   b0     = (int)floorf(c) - EXT;
    float t0     = (float)b0 - c;
    float e_prev = erff(t0 * INV_SQRT2);
#pragma unroll
    for (int o = 0; o < W; ++o) {
        float e_next = erff((t0 + (float)(o + 1)) * INV_SQRT2);
        int   b      = b0 + o;
        float gg     = 0.5f * (e_next - e_prev);
        g[o]         = (b >= 0 && b < nb) ? gg : 0.0f;
        int bcl      = b < 0 ? 0 : b;
        bc[o]        = bcl > nb - 1 ? nb - 1 : bcl;
        e_prev       = e_next;
    }
}

// ---------------------------------------------------------------------------
// Kernel 0: zero demand map (+pad) and the full output vector.
// ---------------------------------------------------------------------------
__global__ void k_zero(float* __restrict__ dem, float* __restrict__ out,
                       int demN, int outN) {
    int i      = blockIdx.x * blockDim.x + threadIdx.x;
    int stride = gridDim.x * blockDim.x;
    for (int j = i; j < demN; j += stride) dem[j] = 0.0f;
    for (int j = i; j < outN; j += stride) out[j] = 0.0f;
}

// ---------------------------------------------------------------------------
// Kernel 1: scatter area*w into the demand map (L2-resident, float atomics).
// ---------------------------------------------------------------------------
__global__ void k_scatter(const float* __restrict__ pos,
                          const float* __restrict__ nsx_,
                          const float* __restrict__ nsy_,
                          const int*   __restrict__ lut_idx,
                          const int*   __restrict__ lut_type,
                          float*       __restrict__ dem,
                          int nInst, int nNodes) {
    int i = blockIdx.x * blockDim.x + threadIdx.x;
    if (i >= nInst) return;
    int   idx  = lut_idx[i];
    float nsx  = nsx_[idx];
    float nsy  = nsy_[idx];
    float cx   = pos[idx] + 0.5f * nsx;
    float cy   = pos[nNodes + idx] + 0.5f * nsy;
    float area = nsx * nsy;
    int   lt   = lut_type[idx];

    float gX[W], gY[W];
    int   bxc[W], byc[W];
    axis_demand(cx, NBX, gX, bxc);
    axis_demand(cy, NBY, gY, byc);

#pragma unroll
    for (int ox = 0; ox < W; ++ox) {
        float ax    = area * gX[ox];
        int   baseX = bxc[ox] * NBY;
#pragma unroll
        for (int oy = 0; oy < W; ++oy) {
            float wv = ax * gY[oy];
            if (wv != 0.0f)
                atomicAdd(&dem[(baseX + byc[oy]) * NBL + lt], wv);
        }
    }
}

// ---------------------------------------------------------------------------
// Kernel 2: compat[c,t] = sum_l dem[c,l] * frac[t,l]  via V_WMMA_F32_16X16X4_F32.
// One wave handles M=16 cells; K=6 is split into two K=4 WMMAs (rows 6,7 of B
// are zeroed so the padded A-operand garbage contributes nothing); N=16 with
// only t<6 written back.
//   A layout: lane m = lane&15 holds row m; component v holds k = v + 2*(lane>>4)
//   B layout: component v holds row k = v + 2*(lane>>4), column n = lane&15
//   D layout: component v holds row (v + 8*(lane>>4)), column n = lane&15
// ---------------------------------------------------------------------------
__global__ void k_compat(const float* __restrict__ dem,
                         const int*   __restrict__ frac,
                         float*       __restrict__ compat) {
    __shared__ float fracLds[NBL * NBL];
    int tid = threadIdx.x;
    if (tid < NBL * NBL) fracLds[tid] = (float)frac[tid];
    __syncthreads();

    int wave = tid >> 5;
    int lane = tid & 31;
    int half = lane >> 4;
    int m    = lane & 15;
    int tileCell = (blockIdx.x * 8 + wave) * 16;   // 16 cells per wave

    const float* dcell = dem + (size_t)(tileCell + m) * NBL;

    // A operands (unconditional loads; k=6,7 hits the zero pad / neighbor but
    // is multiplied by B rows forced to zero below).
    v2f a0, a1;
    a0.x = dcell[2 * half + 0];      // k = 0 / 2
    a0.y = dcell[2 * half + 1];      // k = 1 / 3
    a1.x = dcell[4 + 2 * half + 0];  // k = 4 / 6(dead)
    a1.y = dcell[4 + 2 * half + 1];  // k = 5 / 7(dead)

    // B operands from LDS, masked without divergent memory ops.
    int   mm    = (m < NBL) ? m : 0;
    float maskT = (m < NBL) ? 1.0f : 0.0f;
    float mask1 = (m < NBL && half == 0) ? 1.0f : 0.0f;
    int   l1    = half ? 0 : 4;      // clamped index for dead rows
    v2f b0, b1;
    b0.x = maskT * fracLds[mm * NBL + 2 * half + 0];
    b0.y = maskT * fracLds[mm * NBL + 2 * half + 1];
    b1.x = mask1 * fracLds[mm * NBL + l1 + 0];
    b1.y = mask1 * fracLds[mm * NBL + l1 + 1];

    v8f c = {};
    c = __builtin_amdgcn_wmma_f32_16x16x4_f32(false, a0, false, b0,
                                              (short)0, c, false, false);
    c = __builtin_amdgcn_wmma_f32_16x16x4_f32(false, a1, false, b1,
                                              (short)0, c, false, false);

    if (m < NBL) {
#pragma unroll
        for (int v = 0; v < 8; ++v) {
            int cellRow = tileCell + v + 8 * half;
            compat[(size_t)cellRow * NBL + m] = c[v];
        }
    }
}

// ---------------------------------------------------------------------------
// Kernel 3: gather compat back through each instance footprint.
// ---------------------------------------------------------------------------
__global__ void k_gather(const float* __restrict__ pos,
                         const float* __restrict__ nsx_,
                         const float* __restrict__ nsy_,
                         const int*   __restrict__ lut_idx,
                         const int*   __restrict__ lut_type,
                         const float* __restrict__ compat,
                         float*       __restrict__ out,
                         int nInst, int nNodes) {
    int i = blockIdx.x * blockDim.x + threadIdx.x;
    if (i >= nInst) return;
    int   idx = lut_idx[i];
    float nsx = nsx_[idx];
    float nsy = nsy_[idx];
    float cx  = pos[idx] + 0.5f * nsx;
    float cy  = pos[nNodes + idx] + 0.5f * nsy;
    int   lt  = lut_type[idx];

    float gX[W], gY[W];
    int   bxc[W], byc[W];
    axis_demand(cx, NBX, gX, bxc);
    axis_demand(cy, NBY, gY, byc);

    float acc = 0.0f;
#pragma unroll
    for (int ox = 0; ox < W; ++ox) {
        int baseX = bxc[ox] * NBY;
#pragma unroll
        for (int oy = 0; oy < W; ++oy) {
            float wv = gX[ox] * gY[oy];
            acc += wv * compat[(baseX + byc[oy]) * NBL + lt];
        }
    }
    // / STDXY (=1) / SLICE_CAPACITY (=16)
    out[idx] = acc * 0.0625f;
}

// ---------------------------------------------------------------------------
extern "C" void kernel_launch(void* const* d_in, const int* in_sizes, int n_in,
                              void* d_out, int out_size, void* d_ws, size_t ws_size,
                              hipStream_t stream) {
    const float* pos   = (const float*)d_in[0];
    const float* nsx   = (const float*)d_in[1];
    const float* nsy   = (const float*)d_in[2];
    const int*   lidx  = (const int*)d_in[3];
    const int*   ltype = (const int*)d_in[4];
    const int*   frac  = (const int*)d_in[5];
    float*       out   = (float*)d_out;

    int nNodes = in_sizes[1];   // N = 2,000,000
    int nInst  = in_sizes[3];   // L = 1,000,000

    float* dem    = (float*)d_ws;                 // DEM_N + DEM_PAD floats
    float* compat = dem + DEM_N + DEM_PAD;        // DEM_N floats

    k_zero<<<512, 256, 0, stream>>>(dem, out, DEM_N + DEM_PAD, out_size);
    k_scatter<<<(nInst + 255) / 256, 256, 0, stream>>>(pos, nsx, nsy, lidx,
                                                       ltype, dem, nInst, nNodes);
    k_compat<<<CELLS / 128, 256, 0, stream>>>(dem, frac, compat);  // 630 blocks
    k_gather<<<(nInst + 255) / 256, 256, 0, stream>>>(pos, nsx, nsy, lidx,
                                                      ltype, compat, out,
                                                      nInst, nNodes);
}